// SageGNNClassifier_65859028516894
// MI455X (gfx1250) — compile-verified
//
#include <hip/hip_runtime.h>
#include <hip/hip_bf16.h>
#include <math.h>

typedef __attribute__((ext_vector_type(2))) float v2f;
typedef __attribute__((ext_vector_type(8))) float v8f;

// ---------------------------------------------------------------------------
// Degree computation: deg[dst] += 1 per edge
// ---------------------------------------------------------------------------
__global__ void degree_kernel(const int* __restrict__ dst, float* __restrict__ deg, int E) {
    int e = blockIdx.x * blockDim.x + threadIdx.x;
    if (e >= E) return;
    atomicAdd(&deg[dst[e]], 1.0f);
}

// in-place: deg[i] = 1 / max(deg[i], 1)
__global__ void invdeg_kernel(float* __restrict__ deg, int N) {
    int i = blockIdx.x * blockDim.x + threadIdx.x;
    if (i >= N) return;
    deg[i] = 1.0f / fmaxf(deg[i], 1.0f);
}

// ---------------------------------------------------------------------------
// Edge scatter: agg[dst][c] += x[src][c].
// One thread per (edge, 4-channel chunk): global_load_b128 + 4x
// global_atomic_add_f32.  Consecutive threads own consecutive 16B chunks of
// one edge -> fully coalesced; src/dst index loads amortized 4x.
// ---------------------------------------------------------------------------
template <int C>
__global__ __launch_bounds__(256)
void scatter_add_kernel(const float* __restrict__ x,
                        const int* __restrict__ src,
                        const int* __restrict__ dst,
                        float* __restrict__ agg, int E) {
    constexpr int CHUNKS = C / 4;  // float4 chunks per edge
    int tid = blockIdx.x * blockDim.x + threadIdx.x;
    if (tid >= E * CHUNKS) return;
    int e = tid / CHUNKS;
    int q = tid - e * CHUNKS;
    int s = src[e];
    int d = dst[e];
    const float4 v = *(const float4*)(x + (size_t)s * C + q * 4);
    float* a = agg + (size_t)d * C + q * 4;
    atomicAdd(a + 0, v.x);
    atomicAdd(a + 1, v.y);
    atomicAdd(a + 2, v.z);
    atomicAdd(a + 3, v.w);
}

// ---------------------------------------------------------------------------
// Dense SAGE update via v_wmma_f32_16x16x4_f32:
//   out[i, :] = act( (agg[i,:]*inv[i]) @ Wl^T  +  xin[i,:] @ Wr^T  +  b )
// One wave handles a 16-node tile. A = 16x4 f32 fragment (2 VGPRs),
// B = 4x16 slice of W^T, C/D = 16x16 f32 accumulator (8 VGPRs).
// ---------------------------------------------------------------------------
template <int CIN, int COUT, bool RELU>
__global__ __launch_bounds__(256)
void sage_wmma_kernel(const float* __restrict__ agg,
                      const float* __restrict__ invdeg,
                      const float* __restrict__ xin,
                      const float* __restrict__ wl,   // [COUT][CIN] row-major
                      const float* __restrict__ wr,   // [COUT][CIN] row-major
                      const float* __restrict__ bias, // [COUT]
                      float* __restrict__ out,        // [N][COUT]
                      int N) {
    const int lane = threadIdx.x & 31;
    const int waveInBlock = threadIdx.x >> 5;
    const int tile = blockIdx.x * (blockDim.x >> 5) + waveInBlock;
    const int row0 = tile * 16;
    if (row0 >= N) return;  // wave-uniform exit: EXEC stays all-ones for WMMA

    const int m  = lane & 15;          // matrix row (A) / col (B,D) owned by lane
    const int kk = (lane >> 4) << 1;   // K sub-offset: 0 for lanes 0-15, 2 for 16-31
    const bool fullTile = (row0 + 16 <= N);        // wave-uniform
    const int node = row0 + (fullTile ? m : min(m, N - 1 - row0));
    const float scale = invdeg[node];
    const float* aggRow = agg + (size_t)node * CIN;
    const float* xRow   = xin + (size_t)node * CIN;
    const int rbase = row0 + ((lane >> 4) << 3);   // D rows: v + (lane>=16 ? 8 : 0)

    for (int nb = 0; nb < COUT; nb += 16) {
        v8f c = {};
        const float* wlRow = wl + (size_t)(nb + m) * CIN;
        const float* wrRow = wr + (size_t)(nb + m) * CIN;
#pragma unroll
        for (int k0 = 0; k0 < CIN; k0 += 4) {
            v2f a_mean, a_self, b_l, b_r;
            // A fragment (16x4): lane<16 holds K={k0,k0+1}, lane>=16 K={k0+2,k0+3}
            a_mean.x = aggRow[k0 + kk] * scale;
            a_mean.y = aggRow[k0 + kk + 1] * scale;
            a_self.x = xRow[k0 + kk];
            a_self.y = xRow[k0 + kk + 1];
            // B fragment (4x16 of W^T): VGPR0 = K rows {k0, k0+2}, VGPR1 = {k0+1, k0+3}
            b_l.x = wlRow[k0 + kk];
            b_l.y = wlRow[k0 + kk + 1];
            b_r.x = wrRow[k0 + kk];
            b_r.y = wrRow[k0 + kk + 1];
            c = __builtin_amdgcn_wmma_f32_16x16x4_f32(false, a_mean, false, b_l,
                                                      (short)0, c, false, false);
            c = __builtin_amdgcn_wmma_f32_16x16x4_f32(false, a_self, false, b_r,
                                                      (short)0, c, false, false);
        }
        // D layout: VGPR v -> row v (lanes 0-15) / row v+8 (lanes 16-31), col = m
        const float bv = bias[nb + m];
        if (fullTile) {
            // branch-free store path (the only path taken when N % 16 == 0)
            float* o = out + (size_t)rbase * COUT + nb + m;
#pragma unroll
            for (int v = 0; v < 8; ++v) {
                float val = c[v] + bv;
                if (RELU) val = fmaxf(val, 0.0f);
                o[(size_t)v * COUT] = val;
            }
        } else {
#pragma unroll
            for (int v = 0; v < 8; ++v) {
                int r = rbase + v;
                if (r < N) {
                    float val = c[v] + bv;
                    if (RELU) val = fmaxf(val, 0.0f);
                    out[(size_t)r * COUT + nb + m] = val;
                }
            }
        }
    }
}

// ---------------------------------------------------------------------------
// Final layer: 64 -> 1, sigmoid.  out[i] = sigmoid(mean3[i].wl3 + h2[i].wr3 + b3)
// ---------------------------------------------------------------------------
__global__ void sage_final_kernel(const float* __restrict__ agg,
                                  const float* __restrict__ invdeg,
                                  const float* __restrict__ h2,
                                  const float* __restrict__ wl3,  // [64]
                                  const float* __restrict__ wr3,  // [64]
                                  const float* __restrict__ b3,   // [1]
                                  float* __restrict__ out, int N) {
    int i = blockIdx.x * blockDim.x + threadIdx.x;
    if (i >= N) return;
    float s = b3[0];
    float sc = invdeg[i];
    const float4* ar = (const float4*)(agg + (size_t)i * 64);
    const float4* hr = (const float4*)(h2 + (size_t)i * 64);
    const float4* wlv = (const float4*)wl3;
    const float4* wrv = (const float4*)wr3;
#pragma unroll
    for (int q = 0; q < 16; ++q) {
        float4 a = ar[q], h = hr[q], l = wlv[q], r = wrv[q];
        s += (a.x * sc) * l.x + (a.y * sc) * l.y + (a.z * sc) * l.z + (a.w * sc) * l.w;
        s += h.x * r.x + h.y * r.y + h.z * r.z + h.w * r.w;
    }
    out[i] = 1.0f / (1.0f + expf(-s));
}

// ---------------------------------------------------------------------------
// Launch
// ---------------------------------------------------------------------------
extern "C" void kernel_launch(void* const* d_in, const int* in_sizes, int n_in,
                              void* d_out, int out_size, void* d_ws, size_t ws_size,
                              hipStream_t stream) {
    const float* x    = (const float*)d_in[0];
    const int*   ei   = (const int*)d_in[1];   // [2, E] (src row, dst row)
    const float* wl1  = (const float*)d_in[2];
    const float* wr1  = (const float*)d_in[3];
    const float* b1   = (const float*)d_in[4];
    const float* wl2  = (const float*)d_in[5];
    const float* wr2  = (const float*)d_in[6];
    const float* b2   = (const float*)d_in[7];
    const float* wl3  = (const float*)d_in[8];
    const float* wr3  = (const float*)d_in[9];
    const float* b3   = (const float*)d_in[10];
    float* out = (float*)d_out;

    const int N = in_sizes[0] / 32;   // 100000
    const int E = in_sizes[1] / 2;    // 1600000
    const int* src = ei;
    const int* dst = ei + E;

    // workspace carve: inv-degree [N], agg [N*64], h1 [N*64], h2 [N*64]
    float* deg = (float*)d_ws;
    float* agg = deg + N;
    float* h1  = agg + (size_t)N * 64;
    float* h2  = h1 + (size_t)N * 64;

    const int tiles = (N + 15) / 16;
    const int gemmBlocks = (tiles + 7) / 8;   // 8 waves (16-node tiles) per 256-thr block

    // degrees -> inverse degrees (shared by all 3 layers)
    hipMemsetAsync(deg, 0, (size_t)N * sizeof(float), stream);
    degree_kernel<<<(E + 255) / 256, 256, 0, stream>>>(dst, deg, E);
    invdeg_kernel<<<(N + 255) / 256, 256, 0, stream>>>(deg, N);

    // ---- layer 1: 32 -> 64, ReLU ----
    hipMemsetAsync(agg, 0, (size_t)N * 32 * sizeof(float), stream);
    scatter_add_kernel<32><<<(E * 8 + 255) / 256, 256, 0, stream>>>(x, src, dst, agg, E);
    sage_wmma_kernel<32, 64, true><<<gemmBlocks, 256, 0, stream>>>(
        agg, deg, x, wl1, wr1, b1, h1, N);

    // ---- layer 2: 64 -> 64, ReLU ----
    hipMemsetAsync(agg, 0, (size_t)N * 64 * sizeof(float), stream);
    scatter_add_kernel<64><<<(E * 16 + 255) / 256, 256, 0, stream>>>(h1, src, dst, agg, E);
    sage_wmma_kernel<64, 64, true><<<gemmBlocks, 256, 0, stream>>>(
        agg, deg, h1, wl2, wr2, b2, h2, N);

    // ---- layer 3: 64 -> 1, sigmoid ----
    hipMemsetAsync(agg, 0, (size_t)N * 64 * sizeof(float), stream);
    scatter_add_kernel<64><<<(E * 16 + 255) / 256, 256, 0, stream>>>(h2, src, dst, agg, E);
    sage_final_kernel<<<(N + 255) / 256, 256, 0, stream>>>(
        agg, deg, h2, wl3, wr3, b3, out, N);
}